// RPN_Loss_81922206204091
// MI455X (gfx1250) — compile-verified
//
#include <hip/hip_runtime.h>

// ---------------- problem constants (from reference) ----------------
#define A_TOTAL   80000      // FH*FW*len(RATIOS) = 200*200*2
#define NGT       256
#define IMGF      3200.0f
#define POS_T     0.7f
#define NEG_T     0.3f
#define NSAMP     256
#define POSBUD    128
#define NBLK      313        // ceil(A_TOTAL / 256)

typedef float v2f __attribute__((ext_vector_type(2)));
typedef float v8f __attribute__((ext_vector_type(8)));

// Bijective 32-bit mixer (murmur3 finalizer + seed xor): distinct key per
// anchor index => well-defined "rank" for the subsample selection.
__device__ __forceinline__ unsigned mix32(unsigned x, unsigned seed) {
  x ^= seed;
  x ^= x >> 16; x *= 0x85ebca6bu;
  x ^= x >> 13; x *= 0xc2b2ae35u;
  x ^= x >> 16;
  return x;
}

// ---------------------------------------------------------------------------
// Kernel 1: per-anchor best-IoU (max + argmax over 256 GTs) and base labels.
// GT corner boxes staged in LDS (ds path) so the 80000x256 inner loop reads
// LDS, not VMEM. Anchors loaded as B128.
// ---------------------------------------------------------------------------
__global__ __launch_bounds__(256)
void anchor_pass(const float4* __restrict__ corner_anchors,
                 const float4* __restrict__ target_bnds,   // centre form
                 int* __restrict__ ind, int* __restrict__ labels) {
  __shared__ float ty1[NGT], tx1[NGT], ty2[NGT], tx2[NGT], tarea[NGT];
  const int tid = threadIdx.x;
  {
    float4 t = target_bnds[tid];          // blockDim == NGT == 256
    ty1[tid] = t.x - 0.5f * t.z;  tx1[tid] = t.y - 0.5f * t.w;
    ty2[tid] = t.x + 0.5f * t.z;  tx2[tid] = t.y + 0.5f * t.w;
    tarea[tid] = t.z * t.w;
  }
  __syncthreads();

  const int a = blockIdx.x * 256 + tid;
  if (a >= A_TOTAL) return;
  float4 an = corner_anchors[a];
  {
    int pf = a + 4096; if (pf >= A_TOTAL) pf = A_TOTAL - 1;
    __builtin_prefetch(&corner_anchors[pf], 0, 0);   // -> global_prefetch_b8
  }
  const bool valid = (an.x >= 0.0f) & (an.y >= 0.0f) &
                     (an.z <= IMGF) & (an.w <= IMGF);
  const float area_a = (an.z - an.x) * (an.w - an.y);

  float best = -1.0f;   // masked-invalid value from the reference
  int   bi   = 0;
  if (valid) {
    for (int n = 0; n < NGT; ++n) {
      float y1 = fmaxf(an.x, ty1[n]);
      float x1 = fmaxf(an.y, tx1[n]);
      float y2 = fminf(an.z, ty2[n]);
      float x2 = fminf(an.w, tx2[n]);
      float inter = fmaxf(y2 - y1, 0.0f) * fmaxf(x2 - x1, 0.0f);
      float iou = inter / (area_a + tarea[n] - inter);
      if (iou > best) { best = iou; bi = n; }   // strict > == first-index argmax
    }
  }
  ind[a] = bi;
  int lab = -1;
  if (valid && best < NEG_T) lab = 0;
  if (valid && best > POS_T) lab = 1;
  labels[a] = lab;
}

// ---------------------------------------------------------------------------
// Kernel 2: per-GT argmax over all anchors (column argmax of masked IoU).
// One block per GT; (value, min-index) tree reduction matches jnp.argmax.
// ---------------------------------------------------------------------------
__global__ __launch_bounds__(256)
void gt_argmax(const float4* __restrict__ corner_anchors,
               const float4* __restrict__ target_bnds,
               int* __restrict__ gt_best) {
  const int n = blockIdx.x, tid = threadIdx.x;
  float4 t = target_bnds[n];
  const float gy1 = t.x - 0.5f * t.z, gx1 = t.y - 0.5f * t.w;
  const float gy2 = t.x + 0.5f * t.z, gx2 = t.y + 0.5f * t.w;
  const float area_t = t.z * t.w;

  float best = -2.0f; int bi = 0x7fffffff;
  for (int a = tid; a < A_TOTAL; a += 256) {
    float4 an = corner_anchors[a];
    bool valid = (an.x >= 0.0f) & (an.y >= 0.0f) &
                 (an.z <= IMGF) & (an.w <= IMGF);
    float area_a = (an.z - an.x) * (an.w - an.y);
    float y1 = fmaxf(an.x, gy1), x1 = fmaxf(an.y, gx1);
    float y2 = fminf(an.z, gy2), x2 = fminf(an.w, gx2);
    float inter = fmaxf(y2 - y1, 0.0f) * fmaxf(x2 - x1, 0.0f);
    float iou = inter / (area_a + area_t - inter);
    iou = valid ? iou : -1.0f;
    if (iou > best) { best = iou; bi = a; }   // ascending a => lowest idx kept
  }
  __shared__ float bv[256];
  __shared__ int   bx[256];
  bv[tid] = best; bx[tid] = bi;
  __syncthreads();
  for (int s = 128; s > 0; s >>= 1) {
    if (tid < s) {
      float v = bv[tid + s]; int i = bx[tid + s];
      if (v > bv[tid] || (v == bv[tid] && i < bx[tid])) { bv[tid] = v; bx[tid] = i; }
    }
    __syncthreads();
  }
  if (tid == 0) gt_best[n] = bx[0];
}

// Kernel 3: labels[gt_best[n]] = 1  (force one positive per GT)
__global__ void force_pos(int* __restrict__ labels,
                          const int* __restrict__ gt_best) {
  labels[gt_best[threadIdx.x]] = 1;   // 256 threads; duplicate writes all = 1
}

// ---------------------------------------------------------------------------
// Kernel 4/6: find key threshold = budget-th smallest mix32 key among
// anchors with labels==L (binary search over key space; keys are distinct).
// pass==0: budget = posBudget, writes scalars[0]=kept, scalars[1]=thresh
// pass==1: budget = NSAMP - scalars[0], writes scalars[2]=thresh
// ---------------------------------------------------------------------------
__global__ __launch_bounds__(256)
void select_thresh(const int* __restrict__ labels, int L, unsigned seed,
                   int posBudget, unsigned* __restrict__ scalars, int pass) {
  __shared__ int red[256];
  const int tid = threadIdx.x;
  const int budget = (pass == 0) ? posBudget : (NSAMP - (int)scalars[0]);

  int c = 0;
  for (int a = tid; a < A_TOTAL; a += 256) c += (labels[a] == L);
  red[tid] = c; __syncthreads();
  for (int s = 128; s > 0; s >>= 1) { if (tid < s) red[tid] += red[tid + s]; __syncthreads(); }
  const int P = red[0]; __syncthreads();

  unsigned thresh = 0xffffffffu;            // keep everything
  if (P > budget) {
    unsigned lo = 0u, hi = 0xffffffffu;     // invariant: count(hi) >= budget
    while (lo < hi) {                       // uniform across block
      unsigned mid = lo + ((hi - lo) >> 1);
      int cc = 0;
      for (int a = tid; a < A_TOTAL; a += 256)
        cc += (labels[a] == L && mix32((unsigned)a, seed) <= mid);
      red[tid] = cc; __syncthreads();
      for (int s = 128; s > 0; s >>= 1) { if (tid < s) red[tid] += red[tid + s]; __syncthreads(); }
      int cnt = red[0]; __syncthreads();
      if (cnt >= budget) hi = mid; else lo = mid + 1u;
    }
    thresh = hi;
  }
  if (tid == 0) {
    if (pass == 0) scalars[0] = (unsigned)(P < budget ? P : budget);  // n_pos_kept
    scalars[1 + pass] = thresh;
  }
}

// Kernel 5/7: demote members whose key exceeds the selection threshold.
__global__ __launch_bounds__(256)
void relabel(int* __restrict__ labels, int L, unsigned seed,
             const unsigned* __restrict__ tp) {
  const unsigned th = *tp;
  const int a = blockIdx.x * 256 + threadIdx.x;
  if (a < A_TOTAL && labels[a] == L && mix32((unsigned)a, seed) > th)
    labels[a] = -1;
}

// ---------------------------------------------------------------------------
// Kernel 8: per-block partial sums of 8 quantities:
// [0..3] = sum |pred - target| (y,x,h,w) over positives
// [4]    = sum -logp[:,1] over positives, [5] = sum -logp[:,0] over negatives
// [6]    = n_pos, [7] = n_neg
// Deterministic LDS tree reductions, one partial row per block.
// ---------------------------------------------------------------------------
__global__ __launch_bounds__(256)
void loss_partials(const float* __restrict__ pred_bnd,
                   const float* __restrict__ pred_cls,
                   const float* __restrict__ target_bnds,
                   const float* __restrict__ centre_anchors,
                   const int* __restrict__ labels,
                   const int* __restrict__ ind,
                   float* __restrict__ partials) {
  const int tid = threadIdx.x;
  const int a = blockIdx.x * 256 + tid;
  float acc[8] = {0.f, 0.f, 0.f, 0.f, 0.f, 0.f, 0.f, 0.f};
  if (a < A_TOTAL) {
    const int lab = labels[a];
    const float pos = (lab == 1) ? 1.0f : 0.0f;
    const float neg = (lab == 0) ? 1.0f : 0.0f;
    // 2-class log_softmax
    const float c0 = pred_cls[a * 2 + 0], c1 = pred_cls[a * 2 + 1];
    const float m = fmaxf(c0, c1);
    const float lse = m + logf(expf(c0 - m) + expf(c1 - m));
    acc[4] = -(c1 - lse) * pos;
    acc[5] = -(c0 - lse) * neg;
    acc[6] = pos;
    acc[7] = neg;
    if (lab == 1) {
      const int g = ind[a];
      const float t0 = target_bnds[g * 4 + 0], t1 = target_bnds[g * 4 + 1];
      const float t2 = target_bnds[g * 4 + 2], t3 = target_bnds[g * 4 + 3];
      const float a0 = centre_anchors[a * 4 + 0], a1 = centre_anchors[a * 4 + 1];
      const float a2 = centre_anchors[a * 4 + 2], a3 = centre_anchors[a * 4 + 3];
      const float ty = (t0 - a0) / a2, tx = (t1 - a1) / a3;
      const float th = logf(t2 / a2),  tw = logf(t3 / a3);
      acc[0] = fabsf(pred_bnd[a * 4 + 0] - ty);
      acc[1] = fabsf(pred_bnd[a * 4 + 1] - tx);
      acc[2] = fabsf(pred_bnd[a * 4 + 2] - th);
      acc[3] = fabsf(pred_bnd[a * 4 + 3] - tw);
    }
  }
  __shared__ float red[256];
  for (int q = 0; q < 8; ++q) {
    red[tid] = acc[q]; __syncthreads();
    for (int s = 128; s > 0; s >>= 1) { if (tid < s) red[tid] += red[tid + s]; __syncthreads(); }
    if (tid == 0) partials[blockIdx.x * 8 + q] = red[0];
    __syncthreads();
  }
}

// ---------------------------------------------------------------------------
// Kernel 9: WMMA final reduce. One wave32.
// D = ones(16x4) x B(4x16) + C accumulates exact f32 column sums; columns
// 0..7 carry the 8 partial quantities (other columns fed zeros). All-ones A
// and sum-over-K make the result invariant to the exact K<->lane-half map.
// Row 0 of D lives in lanes 0..15, VGPR0 -> c[0].
// ---------------------------------------------------------------------------
__global__ __launch_bounds__(32)
void final_reduce(const float* __restrict__ partials, float* __restrict__ out) {
  const int lane = threadIdx.x;
  const int n  = lane & 15;        // column
  const int hb = lane >> 4;        // lane half selects K rows {0,1} vs {2,3}
  const int nq = (n < 8) ? n : 0;
  const float nmask = (n < 8) ? 1.0f : 0.0f;

  v8f c = {0.f, 0.f, 0.f, 0.f, 0.f, 0.f, 0.f, 0.f};
  v2f ones; ones.x = 1.0f; ones.y = 1.0f;   // A = all-ones 16x4 (layout-invariant)

  for (int base = 0; base < NBLK; base += 4) {
    const int r0 = base + 2 * hb;
    const int r1 = r0 + 1;
    const float m0 = (r0 < NBLK) ? nmask : 0.0f;
    const float m1 = (r1 < NBLK) ? nmask : 0.0f;
    const int rr0 = (r0 < NBLK) ? r0 : 0;
    const int rr1 = (r1 < NBLK) ? r1 : 0;
    v2f b;
    b.x = partials[rr0 * 8 + nq] * m0;
    b.y = partials[rr1 * 8 + nq] * m1;
    c = __builtin_amdgcn_wmma_f32_16x16x4_f32(false, ones, false, b,
                                              (short)0, c, false, false);
  }

  __shared__ float s[16];
  if (lane < 16) s[lane] = c[0];   // D[M=0, N=lane]
  __syncthreads();
  if (lane == 0) {
    const float sy = s[0], sx = s[1], sh = s[2], sw = s[3];
    const float spc = s[4], snc = s[5], np = s[6], nn = s[7];
    out[0] = sx / np;    // loss_x
    out[1] = sy / np;    // loss_y
    out[2] = sh / np;    // loss_h
    out[3] = sw / np;    // loss_w
    out[4] = spc / np;   // loss_pos_class
    out[5] = snc / nn;   // loss_neg_class
  }
}

// ---------------------------------------------------------------------------
extern "C" void kernel_launch(void* const* d_in, const int* in_sizes, int n_in,
                              void* d_out, int out_size, void* d_ws, size_t ws_size,
                              hipStream_t stream) {
  (void)in_sizes; (void)n_in; (void)out_size; (void)ws_size;

  const float*  pred_bnd = (const float*)d_in[0];   // [1, A, 4]
  const float*  pred_cls = (const float*)d_in[1];   // [1, A, 2]
  const float*  tb       = (const float*)d_in[2];   // [NGT, 4] centre form
  const float4* tb4      = (const float4*)d_in[2];
  const float*  ca       = (const float*)d_in[3];   // centre anchors [A,4]
  const float4* corner   = (const float4*)d_in[4];  // corner anchors [A,4]
  float* out = (float*)d_out;

  char* ws = (char*)d_ws;
  int*      ind      = (int*)ws;                    // A_TOTAL ints   (320000 B)
  int*      labels   = (int*)(ws + 320000);         // A_TOTAL ints   (320000 B)
  int*      gt_best  = (int*)(ws + 640000);         // NGT ints       (1024 B)
  unsigned* scalars  = (unsigned*)(ws + 641024);    // [0]=pos_kept [1]=posTh [2]=negTh
  float*    partials = (float*)(ws + 641056);       // NBLK*8 floats  (10016 B)

  const unsigned SEED_POS = 0x9E3779B9u;
  const unsigned SEED_NEG = 0x7F4A7C15u;

  hipLaunchKernelGGL(anchor_pass, dim3(NBLK), dim3(256), 0, stream,
                     corner, tb4, ind, labels);
  hipLaunchKernelGGL(gt_argmax, dim3(NGT), dim3(256), 0, stream,
                     corner, tb4, gt_best);
  hipLaunchKernelGGL(force_pos, dim3(1), dim3(NGT), 0, stream, labels, gt_best);
  hipLaunchKernelGGL(select_thresh, dim3(1), dim3(256), 0, stream,
                     labels, 1, SEED_POS, POSBUD, scalars, 0);
  hipLaunchKernelGGL(relabel, dim3(NBLK), dim3(256), 0, stream,
                     labels, 1, SEED_POS, scalars + 1);
  hipLaunchKernelGGL(select_thresh, dim3(1), dim3(256), 0, stream,
                     labels, 0, SEED_NEG, 0, scalars, 1);
  hipLaunchKernelGGL(relabel, dim3(NBLK), dim3(256), 0, stream,
                     labels, 0, SEED_NEG, scalars + 2);
  hipLaunchKernelGGL(loss_partials, dim3(NBLK), dim3(256), 0, stream,
                     pred_bnd, pred_cls, tb, ca, labels, ind, partials);
  hipLaunchKernelGGL(final_reduce, dim3(1), dim3(32), 0, stream, partials, out);
}